// FSM_47553877901592
// MI455X (gfx1250) — compile-verified
//
#include <hip/hip_runtime.h>

typedef __attribute__((ext_vector_type(2))) float v2f;
typedef __attribute__((ext_vector_type(8))) float v8f;

#define HW 4096
#define IMW 64

// 3x3 conv, pad=1, + bias + ReLU (+ optional residual add after ReLU).
// One wave32 computes a 16(out-ch) x 16(pixels) tile via V_WMMA_F32_16X16X4_F32.
// Loop nest: 9 filter taps fully unrolled (spatial offset + pad predicate are per-tap,
// channel-invariant), inner loop reduces input channels 4 at a time (WMMA K=4).
// Padding handled with clamped loads + v_cndmask so EXEC stays all-ones.
// CIN: input channels, MT: number of 16-row output-channel tiles (Cout = MT*16).
template <int CIN, int MT, bool RES>
__global__ __launch_bounds__(256) void k_conv3x3(const float* __restrict__ in,
                                                 const float* __restrict__ wgt,
                                                 const float* __restrict__ bias,
                                                 const float* __restrict__ resid,
                                                 float* __restrict__ out) {
  constexpr int K = CIN * 9;
  const int lane = threadIdx.x & 31;
  const int wid  = blockIdx.x * 8 + (threadIdx.x >> 5);
  const int grp  = lane >> 4;   // 0: K={0,1}, 1: K={2,3} (A & B halves)
  const int lg   = lane & 15;   // A: row M, B: col N
  const int b    = wid / (MT * 256);
  const int rem  = wid - b * (MT * 256);
  const int mt   = rem >> 8;
  const int tile = rem & 255;
  const int ty   = tile >> 2;                  // image row
  const int txn  = ((tile & 3) << 4) + lg;     // this lane's pixel column

  const float* inb  = in + (long)b * CIN * HW;       // uniform base (SADDR)
  const int    arow = (mt * 16 + lg) * K + grp * 18; // A: row M, K-half select

  v8f acc = {};
  #pragma unroll
  for (int ky = 0; ky < 3; ++ky) {
    const int  iy  = ty + ky - 1;
    const bool yok = (unsigned)iy < 64u;             // uniform over wave
    #pragma unroll
    for (int kx = 0; kx < 3; ++kx) {
      const int  ix    = txn + kx - 1;
      const bool ok    = yok && ((unsigned)ix < 64u);
      const int  spoff = ok ? ((iy << 6) + ix) : 0;  // clamped, always in-bounds
      const int  tap   = ky * 3 + kx;

      int aoff = arow + tap;           // floats into wgt
      int boff = grp * 2 * HW + spoff; // floats into inb
      for (int c0 = 0; c0 < CIN; c0 += 4) {
        v2f av, bv;
        av.x = wgt[aoff];              // w[m][c0+grp*2+0][tap]
        av.y = wgt[aoff + 9];          // w[m][c0+grp*2+1][tap]
        const float b0 = inb[boff];
        const float b1 = inb[boff + HW];
        bv.x = ok ? b0 : 0.0f;
        bv.y = ok ? b1 : 0.0f;
        // D(16x16,f32) += A(16x4,f32) x B(4x16,f32)
        acc = __builtin_amdgcn_wmma_f32_16x16x4_f32(
            /*neg_a=*/false, av, /*neg_b=*/false, bv,
            /*c_mod=*/(short)0, acc, /*reuse_a=*/false, /*reuse_b=*/false);
        aoff += 36;       // 4 channels * 9 taps
        boff += 4 * HW;   // 4 channel planes
      }
    }
  }

  const int sp = (ty << 6) + txn;
  #pragma unroll
  for (int v = 0; v < 8; ++v) {
    const int m = mt * 16 + grp * 8 + v;   // C/D layout: VGPR v, lane group -> M
    float r = fmaxf(acc[v] + bias[m], 0.0f);
    const long o = (long)b * (MT * 16) * HW + (long)m * HW + sp;
    if constexpr (RES) r += resid[o];
    out[o] = r;
  }
}

// theta/phi/g: three 1x1 convs 16->8 + ReLU, one thread per (batch, pixel).
__global__ __launch_bounds__(256) void k_tpg(const float* __restrict__ x,
    const float* __restrict__ wth, const float* __restrict__ bth,
    const float* __restrict__ wph, const float* __restrict__ bph,
    const float* __restrict__ wg,  const float* __restrict__ bg,
    float* __restrict__ th, float* __restrict__ ph, float* __restrict__ g) {
  const int idx = blockIdx.x * 256 + threadIdx.x;
  const int b = idx >> 12, p = idx & 4095;
  const float* xb = x + b * 16 * HW + p;
  float xv[16];
  #pragma unroll
  for (int c = 0; c < 16; ++c) xv[c] = xb[c * HW];
  #pragma unroll
  for (int j = 0; j < 8; ++j) {
    float s0 = bth[j], s1 = bph[j], s2 = bg[j];
    #pragma unroll
    for (int c = 0; c < 16; ++c) {
      s0 = fmaf(wth[j * 16 + c], xv[c], s0);
      s1 = fmaf(wph[j * 16 + c], xv[c], s1);
      s2 = fmaf(wg [j * 16 + c], xv[c], s2);
    }
    th[b * 8 * HW + j * HW + p] = fmaxf(s0, 0.0f);
    ph[b * 8 * HW + j * HW + p] = fmaxf(s1, 0.0f);
    g [b * 8 * HW + j * HW + p] = fmaxf(s2, 0.0f);
  }
}

// M[b] = phi[b] (8 x HW)  x  g[b]^T (HW x 8): 8x8 Gram matrix per batch.
// This is the associativity rewrite: fg = (theta*phi)*g == theta*(phi*g),
// so the 4096x4096 attention matrix f is never materialized.
__global__ __launch_bounds__(256) void k_phig(const float* __restrict__ ph,
                                              const float* __restrict__ g,
                                              float* __restrict__ Mmat) {
  __shared__ float sm[64];
  const int b = blockIdx.x, tid = threadIdx.x;
  if (tid < 64) sm[tid] = 0.0f;
  __syncthreads();
  float acc[64];
  #pragma unroll
  for (int t = 0; t < 64; ++t) acc[t] = 0.0f;
  const float* pb = ph + b * 8 * HW;
  const float* gb = g  + b * 8 * HW;
  for (int p = tid; p < HW; p += 256) {
    float pv[8], gv[8];
    #pragma unroll
    for (int i = 0; i < 8; ++i) { pv[i] = pb[i * HW + p]; gv[i] = gb[i * HW + p]; }
    #pragma unroll
    for (int i = 0; i < 8; ++i)
      #pragma unroll
      for (int j = 0; j < 8; ++j) acc[i * 8 + j] = fmaf(pv[i], gv[j], acc[i * 8 + j]);
  }
  #pragma unroll
  for (int t = 0; t < 64; ++t) atomicAdd(&sm[t], acc[t]);   // ds_add_f32
  __syncthreads();
  if (tid < 64) Mmat[b * 64 + tid] = sm[tid];
}

// fg[b, p, :] = theta_nchw[b, :, p] x M[b]   (stored flat as [b][p*8+j])
__global__ __launch_bounds__(256) void k_fg(const float* __restrict__ th,
                                            const float* __restrict__ Mmat,
                                            float* __restrict__ fg) {
  const int idx = blockIdx.x * 256 + threadIdx.x;
  const int b = idx >> 12, p = idx & 4095;
  float tv[8];
  #pragma unroll
  for (int i = 0; i < 8; ++i) tv[i] = th[b * 8 * HW + i * HW + p];
  const float* Mb = Mmat + b * 64;
  #pragma unroll
  for (int j = 0; j < 8; ++j) {
    float s = 0.0f;
    #pragma unroll
    for (int i = 0; i < 8; ++i) s = fmaf(tv[i], Mb[i * 8 + j], s);
    fg[b * 8 * HW + p * 8 + j] = s;
  }
}

// y = conv1x1(fg-as-NCHW, w_y) + ReLU; z = y + x.
// Torch-faithful raw reshape: fg_nchw[b][c][s] == fg_flat[b][c*HW + s].
__global__ __launch_bounds__(256) void k_yz(const float* __restrict__ fg,
                                            const float* __restrict__ wy,
                                            const float* __restrict__ by,
                                            const float* __restrict__ x,
                                            float* __restrict__ z) {
  const int idx = blockIdx.x * 256 + threadIdx.x;
  const int b = idx >> 12, s = idx & 4095;
  float fc[8];
  #pragma unroll
  for (int c = 0; c < 8; ++c) fc[c] = fg[b * 8 * HW + c * HW + s];
  #pragma unroll
  for (int r = 0; r < 16; ++r) {
    float acc = by[r];
    #pragma unroll
    for (int c = 0; c < 8; ++c) acc = fmaf(wy[r * 8 + c], fc[c], acc);
    const int o = b * 16 * HW + r * HW + s;
    z[o] = fmaxf(acc, 0.0f) + x[o];
  }
}

extern "C" void kernel_launch(void* const* d_in, const int* in_sizes, int n_in,
                              void* d_out, int out_size, void* d_ws, size_t ws_size,
                              hipStream_t stream) {
  const float* origx  = (const float*)d_in[0];
  const float* w_x    = (const float*)d_in[1];
  const float* b_x    = (const float*)d_in[2];
  const float* w_th   = (const float*)d_in[3];
  const float* b_th   = (const float*)d_in[4];
  const float* w_ph   = (const float*)d_in[5];
  const float* b_ph   = (const float*)d_in[6];
  const float* w_g    = (const float*)d_in[7];
  const float* b_g    = (const float*)d_in[8];
  const float* w_y    = (const float*)d_in[9];
  const float* b_y    = (const float*)d_in[10];
  const float* w_last = (const float*)d_in[11];
  const float* b_last = (const float*)d_in[12];
  float* out = (float*)d_out;

  // Workspace layout (floats): ~4.2 MB total
  float* ws   = (float*)d_ws;
  float* xbuf = ws;                 // 4*16*4096 = 262144
  float* th   = xbuf + 262144;      // 4*8*4096  = 131072
  float* ph   = th   + 131072;      // 131072
  float* gb   = ph   + 131072;      // 131072
  float* Mm   = gb   + 131072;      // 4*64      = 256
  float* fg   = Mm   + 256;         // 131072
  float* zb   = fg   + 131072;      // 262144

  // 1) x = ReLU(conv3x3(origx, w_x) + b_x)        [4,16,64,64]  (WMMA)
  k_conv3x3<48, 1, false><<<128, 256, 0, stream>>>(origx, w_x, b_x, nullptr, xbuf);
  // 2) theta/phi/g = ReLU(conv1x1(x) + b)         [4,8,64,64] each
  k_tpg<<<64, 256, 0, stream>>>(xbuf, w_th, b_th, w_ph, b_ph, w_g, b_g, th, ph, gb);
  // 3) M = phi x g^T over spatial                 [4,8,8]   (f never materialized)
  k_phig<<<4, 256, 0, stream>>>(ph, gb, Mm);
  // 4) fg = theta x M                             [4,4096,8]
  k_fg<<<64, 256, 0, stream>>>(th, Mm, fg);
  // 5) z = ReLU(conv1x1(fg_nchw, w_y) + b_y) + x  [4,16,64,64]
  k_yz<<<64, 256, 0, stream>>>(fg, w_y, b_y, xbuf, zb);
  // 6) out = ReLU(conv3x3(z, w_last) + b_last) + origx   [4,48,64,64]  (WMMA)
  k_conv3x3<16, 3, true><<<384, 256, 0, stream>>>(zb, w_last, b_last, origx, out);
}